// LIIFNonParametric_6296422055941
// MI455X (gfx1250) — compile-verified
//
#include <hip/hip_runtime.h>

// LIIF non-parametric temporal decoder, collapsed form.
//
// Math: per pixel, out[t] = relu(tc[t])*a_p + relu(-tc[t])*a_m  (exact; ReLU
// positive-homogeneity collapses the T loop). a_p/a_m come from three 8x8
// matvec+relu stages using channels 0..199 (200..255 are dead).
//
// Memory-bound kernel: ~105 MB read + 32 MB written => ~6us at 23.3 TB/s.
// The [T x 2] x [2 x pixels] output expansion is done on the matrix unit via
// V_WMMA_F32_16X16X4_F32 (K padded with zeros), stores are non-temporal.

typedef float v2f __attribute__((ext_vector_type(2)));
typedef float v8f __attribute__((ext_vector_type(8)));

#define NB    4
#define CTOT  256
#define HDIM  128
#define WDIM  128
#define HW    (HDIM * WDIM)          // 16384
#define TDIM  64

__global__ __launch_bounds__(256)
void liif_collapsed_wmma(const float* __restrict__ x_real,
                         const float* __restrict__ x_imag,
                         const float* __restrict__ t_coord,
                         float* __restrict__ out)
{
    const int branch = blockIdx.y;                    // 0 = real, 1 = imag
    const float* __restrict__ feat = branch ? x_imag : x_real;

    const int tid  = threadIdx.x;
    const int lane = tid & 31;
    const int pix  = blockIdx.x * 256 + tid;          // flat pixel in [0, B*H*W)
    const int b    = pix >> 14;                       // / HW
    const int rem  = pix & (HW - 1);                  // h*W + w

    // channel-strided weight pointer for this pixel (stride HW floats)
    const float* __restrict__ wp0 = feat + (size_t)b * CTOT * HW + rem;
    const float* __restrict__ wp1 = wp0 + (size_t)64  * HW;
    const float* __restrict__ wp2 = wp0 + (size_t)128 * HW;
    const float* __restrict__ wp3 = wp0 + (size_t)192 * HW;

    // ---- stage 0: s0[o] = sum_i W0[o,i]; p = relu(s0), m = relu(-s0) ----
    float p[8], m[8];
    #pragma unroll
    for (int o = 0; o < 8; ++o) {
        float acc = 0.f;
        #pragma unroll
        for (int i = 0; i < 8; ++i)
            acc += wp0[(size_t)(o * 8 + i) * HW];
        p[o] = fmaxf(acc, 0.f);
        m[o] = fmaxf(-acc, 0.f);
    }

    // ---- stage 1: u = relu(W1 @ {p,m}) ----
    float up[8], um[8];
    #pragma unroll
    for (int o = 0; o < 8; ++o) {
        float ap = 0.f, am = 0.f;
        #pragma unroll
        for (int i = 0; i < 8; ++i) {
            float w = wp1[(size_t)(o * 8 + i) * HW];
            ap = fmaf(w, p[i], ap);
            am = fmaf(w, m[i], am);
        }
        up[o] = fmaxf(ap, 0.f);
        um[o] = fmaxf(am, 0.f);
    }

    // ---- stage 2: v = relu(W2 @ u) ----
    float vp[8], vm[8];
    #pragma unroll
    for (int o = 0; o < 8; ++o) {
        float ap = 0.f, am = 0.f;
        #pragma unroll
        for (int i = 0; i < 8; ++i) {
            float w = wp2[(size_t)(o * 8 + i) * HW];
            ap = fmaf(w, up[i], ap);
            am = fmaf(w, um[i], am);
        }
        vp[o] = fmaxf(ap, 0.f);
        vm[o] = fmaxf(am, 0.f);
    }

    // ---- stage 3: a = W3[row 0] . v  (only channels 192..199 needed) ----
    float a_p = 0.f, a_m = 0.f;
    #pragma unroll
    for (int i = 0; i < 8; ++i) {
        float w = wp3[(size_t)i * HW];
        a_p = fmaf(w, vp[i], a_p);
        a_m = fmaf(w, vm[i], a_m);
    }

    // ---- stage 4: outer-product expansion on the matrix pipe ----
    // out_tile[MxN] = A[Mx4] @ B[4xN]:
    //   A[m,0]=relu(tc), A[m,1]=relu(-tc), A[m,2..3]=0   (M = 16 t values)
    //   B[0,n]=a_p[n],   B[1,n]=a_m[n],   B[2..3,n]=0    (N = 16 pixels)
    // A f32 16x4 layout: 2 VGPRs; lanes 0-15 hold K=0/1, lanes 16-31 K=2/3.
    // B 4x16 row-striped: VGPR0 = rows {K0 | K2}, VGPR1 = rows {K1 | K3}.
    const int half      = lane >> 4;         // 0: low-half lanes, 1: high-half
    const int ln        = lane & 15;
    const int wave_rem0 = rem - lane;        // first pixel of this wave
    float* __restrict__ obase =
        out + (size_t)(branch * NB + b) * TDIM * HW;

    #pragma unroll
    for (int g = 0; g < 2; ++g) {            // two 16-pixel groups per wave
        float bp = __shfl(a_p, g * 16 + ln, 32);
        float bm = __shfl(a_m, g * 16 + ln, 32);
        v2f Bm;
        Bm.x = half ? 0.f : bp;              // K=0 row / K=2 row (zero pad)
        Bm.y = half ? 0.f : bm;              // K=1 row / K=3 row (zero pad)

        #pragma unroll
        for (int c = 0; c < 4; ++c) {        // four 16-row t chunks
            float tc = t_coord[c * 16 + ln];
            v2f Am;
            Am.x = half ? 0.f : fmaxf(tc, 0.f);    // K=0 / K=2(zero)
            Am.y = half ? 0.f : fmaxf(-tc, 0.f);   // K=1 / K=3(zero)

            v8f Cm = {0.f, 0.f, 0.f, 0.f, 0.f, 0.f, 0.f, 0.f};
            v8f D = __builtin_amdgcn_wmma_f32_16x16x4_f32(
                        /*neg_a=*/false, Am,
                        /*neg_b=*/false, Bm,
                        /*c_mod=*/(short)0, Cm,
                        /*reuse_a=*/false, /*reuse_b=*/false);

            // D VGPR v, lane l -> element (t = c*16 + v + 8*half, px = g*16+ln)
            float* __restrict__ op =
                obase + (size_t)(c * 16 + 8 * half) * HW + (wave_rem0 + g * 16 + ln);
            #pragma unroll
            for (int v = 0; v < 8; ++v)
                __builtin_nontemporal_store(D[v], op + (size_t)v * HW);
        }
    }
}

extern "C" void kernel_launch(void* const* d_in, const int* in_sizes, int n_in,
                              void* d_out, int out_size, void* d_ws, size_t ws_size,
                              hipStream_t stream) {
    const float* x_real  = (const float*)d_in[0];
    const float* x_imag  = (const float*)d_in[1];
    const float* t_coord = (const float*)d_in[2];
    float* out = (float*)d_out;

    (void)in_sizes; (void)n_in; (void)out_size; (void)d_ws; (void)ws_size;

    dim3 grid((NB * HW) / 256, 2, 1);   // 256 blocks x {real,imag}
    dim3 block(256, 1, 1);
    liif_collapsed_wmma<<<grid, block, 0, stream>>>(x_real, x_imag, t_coord, out);
}